// HiPoolLarge_12343736009145
// MI455X (gfx1250) — compile-verified
//
#include <hip/hip_runtime.h>
#include <math.h>

typedef float v2f __attribute__((ext_vector_type(2)));
typedef float v8f __attribute__((ext_vector_type(8)));

#define N_NODES 4096
#define NDIM    128
#define N1      2048

// ---------------------------------------------------------------------------
// fp32 WMMA GEMM, fully compile-time specialized:
//   one wave32 computes a 16 x (16*NB) strip of C using V_WMMA_F32_16X16X4_F32,
//   K stepped by 4, A fragment reused across the NB column tiles.
// A row-major [M,KK] (ld LDA). B row-major [KK,N] (BT=false, ld LDB) or stored
// [N,KK] row-major = logical B^T (BT=true). C row-major ld LDC.
// ---------------------------------------------------------------------------
enum { EPI_NONE = 0, EPI_MASK = 1, EPI_ADDBASE = 2, EPI_BIAS_RELU = 3, EPI_BIAS = 4 };

template <bool BT, int EPI, int NB, int LDA, int LDB, int LDC, int KK>
__global__ __launch_bounds__(32) void gemm_wmma_f32(
    const float* __restrict__ A,
    const float* __restrict__ B,
    float* __restrict__ C,
    const float* __restrict__ extra)
{
    const int lane   = threadIdx.x;       // wave32
    const int mn     = lane & 15;         // A: row m   B: col n (within tile)
    const int kh     = (lane >> 4) << 1;  // k sub-offset: 0 or 2
    const int tileM  = blockIdx.y << 4;
    const int tileN0 = blockIdx.x * (16 * NB);

    const float* Ap = A + (tileM + mn) * LDA + kh;

    v8f acc[NB];
#pragma unroll
    for (int t = 0; t < NB; ++t) acc[t] = (v8f){0.f,0.f,0.f,0.f,0.f,0.f,0.f,0.f};

#pragma unroll 4
    for (int k0 = 0; k0 < KK; k0 += 4) {
        // A 16x4 fragment: lanes 0-15 hold k={k0,k0+1}, lanes 16-31 k={k0+2,k0+3}
        const v2f a = *(const v2f*)(Ap + k0);
#pragma unroll
        for (int t = 0; t < NB; ++t) {
            v2f b;
            if constexpr (BT) {
                b = *(const v2f*)(B + (tileN0 + 16 * t + mn) * LDB + kh + k0);
            } else {
                b.x = B[(k0 + kh)     * LDB + tileN0 + 16 * t + mn];
                b.y = B[(k0 + kh + 1) * LDB + tileN0 + 16 * t + mn];
            }
            acc[t] = __builtin_amdgcn_wmma_f32_16x16x4_f32(
                /*neg_a=*/false, a, /*neg_b=*/false, b,
                /*c_mod=*/(short)0, acc[t], /*reuse_a=*/false, /*reuse_b=*/false);
        }
    }

    // C/D layout: VGPR r -> row r (lanes 0-15) / row r+8 (lanes 16-31), col = lane&15
    const int hi = lane >> 4;
#pragma unroll
    for (int t = 0; t < NB; ++t) {
        const int gn = tileN0 + 16 * t + mn;
#pragma unroll
        for (int r = 0; r < 8; ++r) {
            const int gm = tileM + r + (hi << 3);
            float v = acc[t][r];
            if constexpr (EPI == EPI_MASK) {
                // zero where column is a member row of cluster gm: (gn-2*gm) mod 4096 < 4
                int d = (gn + N_NODES - 2 * gm) & (N_NODES - 1);
                if (d < 4) v = 0.0f;            // score*0 (NOT -inf) per reference
            } else if constexpr (EPI == EPI_ADDBASE) {
                v += extra[gm * LDC + gn];
            } else if constexpr (EPI == EPI_BIAS_RELU) {
                v = fmaxf(v + extra[gn], 0.0f);
            } else if constexpr (EPI == EPI_BIAS) {
                v += extra[gn];
            }
            C[gm * LDC + gn] = v;
        }
    }
}

// ---------------------------------------------------------------------------
// flat_s.T @ M  ==  4-row pooling: out[j,:] = sum_{t<4} in[(2j+t) mod rows,:]
// (clusters j >= jmax are all-zero columns of the assignment -> output 0)
// ---------------------------------------------------------------------------
__global__ void pool4_rows(const float* __restrict__ in, float* __restrict__ out,
                           int ncols, int rowmask, int jmax, int total)
{
    int idx = blockIdx.x * blockDim.x + threadIdx.x;
    if (idx >= total) return;
    int j = idx / ncols;
    int c = idx - j * ncols;
    float s = 0.0f;
    if (j < jmax) {
        int r0 = j << 1;
#pragma unroll
        for (int t = 0; t < 4; ++t)
            s += in[(size_t)((r0 + t) & rowmask) * ncols + c];
    }
    out[idx] = s;
}

// M @ flat_s == 4-col pooling: out[:,k] = sum_{t<4} in[:,(2k+t) mod cols]
__global__ void pool4_cols(const float* __restrict__ in, float* __restrict__ out,
                           int ncols_in, int ncols_out, int colmask, int kmax, int total)
{
    int idx = blockIdx.x * blockDim.x + threadIdx.x;
    if (idx >= total) return;
    int j = idx / ncols_out;
    int k = idx - j * ncols_out;
    float s = 0.0f;
    if (k < kmax) {
        int c0 = k << 1;
#pragma unroll
        for (int t = 0; t < 4; ++t)
            s += in[(size_t)j * ncols_in + ((c0 + t) & colmask)];
    }
    out[idx] = s;
}

// Row-wise softmax over ncols (in place)
__global__ void softmax_rows(float* __restrict__ S, int ncols)
{
    const int row = blockIdx.x;
    float* r = S + (size_t)row * ncols;
    __shared__ float red[256];
    const int tid = threadIdx.x;

    float m = -3.402823466e38f;
    for (int c = tid; c < ncols; c += 256) m = fmaxf(m, r[c]);
    red[tid] = m; __syncthreads();
    for (int s = 128; s > 0; s >>= 1) {
        if (tid < s) red[tid] = fmaxf(red[tid], red[tid + s]);
        __syncthreads();
    }
    m = red[0]; __syncthreads();

    float sum = 0.0f;
    for (int c = tid; c < ncols; c += 256) {
        float e = expf(r[c] - m);
        r[c] = e;
        sum += e;
    }
    red[tid] = sum; __syncthreads();
    for (int s = 128; s > 0; s >>= 1) {
        if (tid < s) red[tid] += red[tid + s];
        __syncthreads();
    }
    const float inv = 1.0f / red[0];
    for (int c = tid; c < ncols; c += 256) r[c] *= inv;
}

// d[i] = clip(rowsum with diag forced to 1, >=1)^-1/2
__global__ void degree_rsqrt(const float* __restrict__ adj, float* __restrict__ d, int n)
{
    const int row = blockIdx.x;
    const float* r = adj + (size_t)row * n;
    __shared__ float red[256];
    const int tid = threadIdx.x;
    float s = 0.0f;
    for (int c = tid; c < n; c += 256) s += r[c];
    red[tid] = s; __syncthreads();
    for (int st = 128; st > 0; st >>= 1) {
        if (tid < st) red[tid] += red[tid + st];
        __syncthreads();
    }
    if (tid == 0) {
        float rs = red[0] - r[row] + 1.0f;   // replace diagonal entry with 1
        rs = fmaxf(rs, 1.0f);
        d[row] = rsqrtf(rs);
    }
}

// out[i,j] = d[i] * (i==j ? 1 : adj[i,j]) * d[j]
__global__ void norm_adj(const float* __restrict__ adj, const float* __restrict__ d,
                         float* __restrict__ out, int n, int total)
{
    int idx = blockIdx.x * blockDim.x + threadIdx.x;
    if (idx >= total) return;
    int i = idx / n;
    int j = idx - i * n;
    float a = (i == j) ? 1.0f : adj[idx];
    out[idx] = d[i] * a * d[j];
}

// mean over rows -> [1,H]; relu(mean @ Wlin.T + blin) -> [1,2]; log_softmax
__global__ void final_head(const float* __restrict__ h2, const float* __restrict__ Wlin,
                           const float* __restrict__ blin, float* __restrict__ out)
{
    __shared__ float m[NDIM];
    __shared__ float o[2];
    const int c = threadIdx.x;   // 128 threads
    float s = 0.0f;
    for (int r = 0; r < N1; ++r) s += h2[(size_t)r * NDIM + c];
    m[c] = s * (1.0f / (float)N1);
    __syncthreads();
    if (c < 2) {
        float acc = blin[c];
        for (int k = 0; k < NDIM; ++k) acc += m[k] * Wlin[c * NDIM + k];
        o[c] = fmaxf(acc, 0.0f);
    }
    __syncthreads();
    if (c == 0) {
        float mx  = fmaxf(o[0], o[1]);
        float lse = mx + logf(expf(o[0] - mx) + expf(o[1] - mx));
        out[0] = o[0] - lse;
        out[1] = o[1] - lse;
    }
}

// ---------------------------------------------------------------------------
extern "C" void kernel_launch(void* const* d_in, const int* in_sizes, int n_in,
                              void* d_out, int out_size, void* d_ws, size_t ws_size,
                              hipStream_t stream)
{
    const float* x    = (const float*)d_in[0];
    const float* adj  = (const float*)d_in[1] + (size_t)N_NODES * N_NODES; // edge_index[1]
    const float* ca1  = (const float*)d_in[3];
    const float* W1   = (const float*)d_in[4];
    const float* b1   = (const float*)d_in[5];
    const float* W2   = (const float*)d_in[6];
    const float* b2   = (const float*)d_in[7];
    const float* Wlin = (const float*)d_in[8];
    const float* blin = (const float*)d_in[9];

    // workspace layout (floats); everything fits in MI455X's 192MB L2
    float* ws   = (float*)d_ws;
    float* x1   = ws;                          // [2048,128] x1_base, then final x1
    float* t    = x1   + N1 * NDIM;            // [2048,128] x1@ca1
    float* xw   = t    + N1 * NDIM;            // [2048,128] x1@W1
    float* x2   = xw   + N1 * NDIM;            // [2048,128]
    float* x2w  = x2   + N1 * NDIM;            // [2048,128] x2@W2
    float* h    = x2w  + N1 * NDIM;            // [2048,128] GCN1 out
    float* h2   = h    + N1 * NDIM;            // [2048,128] GCN2 out
    float* d1   = h2   + N1 * NDIM;            // [2048]
    float* d2   = d1   + N1;                   // [2048]
    float* adj1 = d2   + N1;                   // [2048,2048] original pooled adjacency
    float* adjn = adj1 + (size_t)N1 * N1;      // [2048,2048] normalized (reused lvl1/lvl2)
    float* adj2 = adjn + (size_t)N1 * N1;      // [2048,2048]
    float* big  = adj2 + (size_t)N1 * N1;      // [2048,4096] rowpool tmp / scores / tmp2

    const int TB = 256;

    // 1) x1_base = flat_s.T @ x   (4-row pooling, wrap mod 4096)
    pool4_rows<<<(N1 * NDIM + TB - 1) / TB, TB, 0, stream>>>(
        x, x1, NDIM, N_NODES - 1, N1, N1 * NDIM);

    // 2-3) adj1 = flat_s.T @ adj @ flat_s   (row-pool then col-pool)
    pool4_rows<<<(N1 * N_NODES + TB - 1) / TB, TB, 0, stream>>>(
        adj, big, N_NODES, N_NODES - 1, N1, N1 * N_NODES);
    pool4_cols<<<(N1 * N1 + TB - 1) / TB, TB, 0, stream>>>(
        big, adj1, N_NODES, N1, N_NODES - 1, N1, N1 * N1);

    // 4) t = x1 @ ca1            [2048,128]x[128,128]
    gemm_wmma_f32<false, EPI_NONE, 2, NDIM, NDIM, NDIM, NDIM>
        <<<dim3(NDIM / 32, N1 / 16), 32, 0, stream>>>(x1, ca1, t, nullptr);

    // 5) scores = (t @ x.T) * reverse_s.T   [2048,4096], masked-to-zero, NB=4
    gemm_wmma_f32<true, EPI_MASK, 4, NDIM, NDIM, N_NODES, NDIM>
        <<<dim3(N_NODES / 64, N1 / 16), 32, 0, stream>>>(t, x, big, nullptr);

    // 6) alpha = softmax(scores, axis=1)
    softmax_rows<<<N1, TB, 0, stream>>>(big, N_NODES);

    // 7) x1 = alpha @ x + x1_base          [2048,4096]x[4096,128]
    gemm_wmma_f32<false, EPI_ADDBASE, 2, N_NODES, NDIM, NDIM, N_NODES>
        <<<dim3(NDIM / 32, N1 / 16), 32, 0, stream>>>(big, x, x1, x1);

    // 8-9) GCN1 normalization: d1, adjn = D^-1/2 (adj1 + I on diag) D^-1/2
    degree_rsqrt<<<N1, TB, 0, stream>>>(adj1, d1, N1);
    norm_adj<<<(N1 * N1 + TB - 1) / TB, TB, 0, stream>>>(adj1, d1, adjn, N1, N1 * N1);

    // 10) xw = x1 @ W1
    gemm_wmma_f32<false, EPI_NONE, 2, NDIM, NDIM, NDIM, NDIM>
        <<<dim3(NDIM / 32, N1 / 16), 32, 0, stream>>>(x1, W1, xw, nullptr);

    // 11) h = relu(adjn @ xw + b1)         [2048,2048]x[2048,128]
    gemm_wmma_f32<false, EPI_BIAS_RELU, 2, N1, NDIM, NDIM, N1>
        <<<dim3(NDIM / 32, N1 / 16), 32, 0, stream>>>(adjn, xw, h, b1);

    // 12) x2 = flat_s2.T @ h  (4-row pooling mod 2048, clusters >= 1024 are zero)
    pool4_rows<<<(N1 * NDIM + TB - 1) / TB, TB, 0, stream>>>(
        h, x2, NDIM, N1 - 1, N1 / 2, N1 * NDIM);

    // 13-14) adj2 = flat_s2.T @ adj1 @ flat_s2 (original adj1, no diag/norm)
    pool4_rows<<<(N1 * N1 + TB - 1) / TB, TB, 0, stream>>>(
        adj1, big, N1, N1 - 1, N1 / 2, N1 * N1);
    pool4_cols<<<(N1 * N1 + TB - 1) / TB, TB, 0, stream>>>(
        big, adj2, N1, N1, N1 - 1, N1 / 2, N1 * N1);

    // 15) x2w = x2 @ W2
    gemm_wmma_f32<false, EPI_NONE, 2, NDIM, NDIM, NDIM, NDIM>
        <<<dim3(NDIM / 32, N1 / 16), 32, 0, stream>>>(x2, W2, x2w, nullptr);

    // 16-17) GCN2 normalization
    degree_rsqrt<<<N1, TB, 0, stream>>>(adj2, d2, N1);
    norm_adj<<<(N1 * N1 + TB - 1) / TB, TB, 0, stream>>>(adj2, d2, adjn, N1, N1 * N1);

    // 18) h2 = adjn @ x2w + b2  (no relu)
    gemm_wmma_f32<false, EPI_BIAS, 2, N1, NDIM, NDIM, N1>
        <<<dim3(NDIM / 32, N1 / 16), 32, 0, stream>>>(adjn, x2w, h2, b2);

    // 19) mean -> linear -> relu -> log_softmax
    final_head<<<1, NDIM, 0, stream>>>(h2, Wlin, blin, (float*)d_out);

    (void)in_sizes; (void)n_in; (void)out_size; (void)ws_size;
}